// RoIPointPool3d_67937792688494
// MI455X (gfx1250) — compile-verified
//
#include <hip/hip_runtime.h>
#include <hip/hip_bf16.h>
#include <stdint.h>

// ---------------- problem constants (from setup_inputs) ----------------
#define SAMP 512            // NUM_SAMPLED
#define EXTRA_W 1.0f        // EXTRA_WIDTH
static constexpr int Bb = 4, Nn = 16384, Cc = 128, Mm = 128;
static constexpr int CH = 256;              // points per staged chunk
static constexpr int NCHUNK = Nn / CH;      // 64
static constexpr int ROW = 3 + Cc;          // 131 f32 per pooled row

// ---------------- CDNA5 feature detection ----------------
#if __has_builtin(__builtin_amdgcn_tensor_load_to_lds)
#define HAS_TDM 1
#else
#define HAS_TDM 0
#endif

#if __has_builtin(__builtin_amdgcn_s_wait_tensorcnt)
#define WAIT_TENSOR(n) __builtin_amdgcn_s_wait_tensorcnt((short)(n))
#else
#define WAIT_TENSOR(n) asm volatile("s_wait_tensorcnt %0" ::"i"(n))
#endif

#if __has_builtin(__builtin_amdgcn_ballot_w32)
#define BALLOT32(x) __builtin_amdgcn_ballot_w32(x)
#else
#define BALLOT32(x) ((unsigned)__ballot(x))
#endif

typedef unsigned int v4u __attribute__((ext_vector_type(4)));
typedef int          v8i __attribute__((ext_vector_type(8)));
typedef int          v4i __attribute__((ext_vector_type(4)));

#if HAS_TDM
// 1-D contiguous DMA of `ndw` dwords global -> LDS via the Tensor Data Mover.
// D# per CDNA5 ISA ch.8: group0 {count=1, lds_addr, global_addr[56:0], type=2},
// group1 {data_size=4B, tensor_dim0=tile_dim0=ndw, tensor_dim1=tile_dim1=1},
// groups 2/3 describe size-1 upper dims (both-non-NULL 4-group form).
__device__ __forceinline__ void tdm_load_dwords(const void* gsrc,
                                                unsigned lds_off, int ndw) {
  unsigned long long ga = (unsigned long long)(uintptr_t)gsrc;
  v4u g0;
  g0.x = 1u;                                             // count=1, user mode
  g0.y = lds_off;                                        // LDS byte address
  g0.z = (unsigned)ga;                                   // global_addr[31:0]
  g0.w = (unsigned)((ga >> 32) & 0x1FFFFFFull)           // global_addr[56:32]
         | (2u << 30);                                   // type = 2 ("image")
  v8i g1;
  g1[0] = (int)(2u << 16);                               // data_size = 4 bytes
  g1[1] = (int)((unsigned)(ndw & 0xFFFF) << 16);         // tensor_dim0 lo16
  g1[2] = (int)((((unsigned)ndw >> 16) & 0xFFFFu)        // tensor_dim0 hi16
                | (1u << 16));                           // tensor_dim1 = 1
  g1[3] = (int)((unsigned)(ndw & 0xFFFF) << 16);         // tile_dim0 = ndw
  g1[4] = 1;                                             // tile_dim1=1, tile_dim2=0
  g1[5] = ndw;                                           // tensor_dim0_stride lo32
  g1[6] = 0;
  g1[7] = 0;
  v4i g2 = {1, 1, 0, 0};                                 // tensor_dim2=1, dim3=1
  v4i g3 = {0, 0x00010000, 0, 0};                        // tensor_dim4=1
#if defined(__clang_major__) && (__clang_major__ >= 23)
  v8i g4 = {0, 0, 0, 0, 0, 0, 0, 0};
  __builtin_amdgcn_tensor_load_to_lds(g0, g1, g2, g3, g4, 0);
#else
  __builtin_amdgcn_tensor_load_to_lds(g0, g1, g2, g3, 0);
#endif
}
#endif

// One workgroup (256 thr = 8 wave32) per (batch, box).
__global__ __launch_bounds__(256) void roipool3d_kernel(
    const float* __restrict__ points,   // (B,N,3)
    const float* __restrict__ feats,    // (B,N,C)
    const float* __restrict__ boxes,    // (B,M,7)
    float* __restrict__ outF,           // (B,M,SAMP,ROW)
    int* __restrict__ flags)            // (B,M)
{
  __shared__ float buf[2][CH * 3];      // double-buffered xyz chunks (TDM dest)
  __shared__ int   sel[SAMP];           // first SAMP in-box point indices
  __shared__ int   wcnt[8];             // per-wave mask popcounts

  const int bm   = blockIdx.x;
  const int b    = bm / Mm;
  const int t    = threadIdx.x;
  const int lane = t & 31;
  const int wv   = t >> 5;

  const float* pbase = points + (size_t)b * Nn * 3;

#if HAS_TDM
  // Kick off the DMA of chunk 0 first so it overlaps the scalar sincos below.
  if (t < 32)  // wave 0 drives the DMA pipeline (TENSORcnt is per-wave)
    tdm_load_dwords(pbase, (unsigned)(uintptr_t)&buf[0][0], CH * 3);
#endif

  // Box parameters (wave-uniform).
  const float* bx = boxes + (size_t)bm * 7;
  const float cx = bx[0], cy = bx[1], cz = bx[2];
  const float hx = bx[3] * 0.5f + EXTRA_W;
  const float hy = bx[4] * 0.5f + EXTRA_W;
  const float hz = bx[5] * 0.5f + EXTRA_W;
  const float chd = cosf(bx[6]);
  const float shd = sinf(bx[6]);

  // ---- phase 1: in-order stream compaction of in-box point indices ----
  int base = 0;  // selected-so-far (uniform across all threads)
  for (int c = 0; c < NCHUNK; ++c) {
    const int pb = c & 1;
#if HAS_TDM
    if (t < 32) {
      if (c + 1 < NCHUNK) {
        tdm_load_dwords(pbase + (size_t)(c + 1) * CH * 3,
                        (unsigned)(uintptr_t)&buf[pb ^ 1][0], CH * 3);
        WAIT_TENSOR(1);   // chunk c landed; chunk c+1 may be in flight
      } else {
        WAIT_TENSOR(0);
      }
    }
    __syncthreads();
#else
    for (int i = t; i < CH * 3; i += 256)
      buf[pb][i] = pbase[(size_t)c * CH * 3 + i];
    __syncthreads();
#endif
    const float px = buf[pb][3 * t + 0];
    const float py = buf[pb][3 * t + 1];
    const float pz = buf[pb][3 * t + 2];
    const float sx = px - cx, sy = py - cy, sz = pz - cz;
    const float lx = sx * chd + sy * shd;
    const float ly = sy * chd - sx * shd;
    const bool m = (fabsf(lx) < hx) && (fabsf(ly) < hy) && (fabsf(sz) < hz);

    const unsigned bal = BALLOT32(m);
    if (lane == 0) wcnt[wv] = __popc(bal);
    __syncthreads();
    int waveoff = 0, total = 0;
#pragma unroll
    for (int i = 0; i < 8; ++i) {
      const int v = wcnt[i];
      total += v;
      if (i < wv) waveoff += v;
    }
    if (m) {
      const int g = base + waveoff + __popc(bal & ((1u << lane) - 1u));
      if (g < SAMP) sel[g] = c * CH + t;
    }
    base += total;
    __syncthreads();                 // protect wcnt/buf reuse next iteration
    if (base >= SAMP) break;         // uniform early exit; exact cnt no longer needed
  }
#if HAS_TDM
  if (t < 32) WAIT_TENSOR(0);        // drain any in-flight DMA before exit paths
#endif
  __syncthreads();

  const int cnt = base;
  if (t == 0) flags[bm] = (cnt == 0) ? 1 : 0;

  // ---- phase 2: gather / write (one wave per output row) ----
  float* orow0 = outF + (size_t)bm * SAMP * ROW;
  if (cnt == 0) {
    for (int k = wv; k < SAMP; k += 8) {
      float* orow = orow0 + (size_t)k * ROW;
      if (lane < 3) orow[lane] = 0.0f;
      float* o4 = orow + 3 + 4 * lane;
      o4[0] = 0.0f; o4[1] = 0.0f; o4[2] = 0.0f; o4[3] = 0.0f;
    }
    return;
  }
  const int cw = cnt < SAMP ? cnt : SAMP;  // wrap modulus (== cnt when cnt<512)

  // Pre-expand the wrap pattern in place: writers touch only sel[cw..SAMP),
  // readers only sel[0..cw) -> race-free, moves the idiv off the hot loop.
  for (int k = cw + t; k < SAMP; k += 256) sel[k] = sel[k % cw];
  __syncthreads();

  const float* fbase = feats + (size_t)b * Nn * Cc;
  for (int k = wv; k < SAMP; k += 8) {
    const int src = sel[k];                              // wave-uniform broadcast
    const float4 v = ((const float4*)(fbase + (size_t)src * Cc))[lane];
    float* orow = orow0 + (size_t)k * ROW;
    if (lane < 3) orow[lane] = pbase[(size_t)src * 3 + lane];
    float* o4 = orow + 3 + 4 * lane;                     // 4-byte aligned
    o4[0] = v.x; o4[1] = v.y; o4[2] = v.z; o4[3] = v.w;
  }
}

extern "C" void kernel_launch(void* const* d_in, const int* in_sizes, int n_in,
                              void* d_out, int out_size, void* d_ws,
                              size_t ws_size, hipStream_t stream) {
  const float* points = (const float*)d_in[0];  // (4,16384,3)
  const float* feats  = (const float*)d_in[1];  // (4,16384,128)
  const float* boxes  = (const float*)d_in[2];  // (4,128,7)
  float* outF = (float*)d_out;
  int* flags = (int*)(outF + (size_t)Bb * Mm * SAMP * ROW);  // int32 tail
  roipool3d_kernel<<<dim3(Bb * Mm), 256, 0, stream>>>(points, feats, boxes,
                                                      outF, flags);
}